// MultiplicativeContourIntegrationLayer_28424093565383
// MI455X (gfx1250) — compile-verified
//
#include <hip/hip_runtime.h>
#include <stdint.h>

// Problem constants (from reference)
#define BN   64          // batch
#define RR   55          // rows
#define CCOL 55          // cols
#define CHN  96          // channels
#define NK   25          // raw kernel size
#define NT   9           // masked taps per axis (stride-3 lattice)
#define CG   8           // channels staged per workgroup
#define NPIX (RR * CCOL) // 3025
#define NBLK (BN * (CHN / CG))

typedef unsigned int v4u __attribute__((ext_vector_type(4)));
typedef int          v8i __attribute__((ext_vector_type(8)));
typedef int          v4i __attribute__((ext_vector_type(4)));

struct __align__(16) Smem {
  float x[NPIX * CG];      // [row][col][cg]  -- filled by TDM, 96.8 KB
  float k[NT * NT * CG];   // [tap][cg]       -- masked taps, center zeroed
};

__global__ __launch_bounds__(256)
void contour_lateral_kernel(const float* __restrict__ x,
                            const float* __restrict__ rk,
                            const float* __restrict__ bias,
                            float* __restrict__ out) {
  __shared__ Smem sm;

  const int tid = threadIdx.x;
  const int b   = blockIdx.x / (CHN / CG);
  const int ch0 = (blockIdx.x % (CHN / CG)) * CG;

  // ---- stage the 81 masked taps (center zeroed) into LDS: k[tap][cg] ----
  for (int idx = tid; idx < NT * NT * CG; idx += 256) {
    const int tap = idx >> 3;
    const int ch  = idx & (CG - 1);
    const int ti  = tap / NT;
    const int tj  = tap - ti * NT;
    float v = rk[(size_t)(ch0 + ch) * NK * NK + (3 * ti) * NK + (3 * tj)];
    sm.k[idx] = (tap == (NT * NT) / 2) ? 0.0f : v;  // center (i=4,j=4)
  }

  // ---- TDM: one async 3-D tile DMA (CG x 55cols x 55rows) global -> LDS ----
#if __has_builtin(__builtin_amdgcn_tensor_load_to_lds) && __has_builtin(__builtin_amdgcn_s_wait_tensorcnt)
  if ((tid >> 5) == 0) {  // wave 0 only (EXEC ignored by TDM; branch is wave-uniform)
    const uint64_t gaddr =
        (uint64_t)(uintptr_t)(x + (size_t)b * NPIX * CHN + ch0);
    // D# group 0: count=1 | lds_addr=0 (sm.x at LDS base) | global_addr | type=2
    v4u g0 = { 1u,
               0u,
               (unsigned)(gaddr & 0xFFFFFFFFu),
               (unsigned)(((gaddr >> 32) & 0x1FFFFFFu) | (2u << 30)) };
    // D# group 1:
    //  w0: workgroup_mask=0, data_size=2 (4B), no barrier/iterate/pad
    //  dim0 = CG channels (unit stride), dim1 = cols (stride 96), dim2 = rows (stride 5280)
    v8i g1 = { (int)(2u << 16),                              // data_size = 4B
               (int)((CG & 0xFFFF) << 16),                   // tensor_dim0 lo16
               (int)((CG >> 16) | ((CCOL & 0xFFFF) << 16)),  // dim0 hi | tensor_dim1 lo
               (int)((CCOL >> 16) | (CG << 16)),             // dim1 hi | tile_dim0
               (int)(CCOL | (RR << 16)),                     // tile_dim1 | tile_dim2
               (int)CHN,                                     // tensor_dim0_stride lo32 (=96)
               (int)(((CCOL * CHN) & 0xFFFF) << 16),         // stride0 hi16(=0) | stride1 lo16
               (int)((CCOL * CHN) >> 16) };                  // tensor_dim1_stride hi32
    v4i g2 = { (int)RR, 0, 0, 0 };                           // tensor_dim2 = rows
    v4i g3 = { 0, 0, 0, 0 };
    v8i gx = { 0, 0, 0, 0, 0, 0, 0, 0 };                     // unused 5th group (6-arg form)
    __builtin_amdgcn_tensor_load_to_lds(g0, g1, g2, g3, gx, /*cpol=*/0);
    __builtin_amdgcn_s_wait_tensorcnt(0);
  }
#else
  // Fallback: cooperative direct staging (keeps HBM traffic single-touch too)
  for (int idx = tid; idx < NPIX * CG; idx += 256) {
    const int p  = idx >> 3;
    const int ch = idx & (CG - 1);
    sm.x[idx] = x[(size_t)b * NPIX * CHN + (size_t)p * CHN + ch0 + ch];
  }
#endif
  __syncthreads();

  // ---- compute: thread owns 4 channels (b128 LDS traffic), strides pixels ----
  const int ch4 = (tid & 1) * 4;  // which half of the 8-channel group
  const float4 bv = *(const float4*)&bias[ch0 + ch4];
  const size_t obase = (size_t)b * NPIX * CHN + ch0 + ch4;

  for (int p = (tid >> 1); p < NPIX; p += 128) {
    const int r = p / CCOL;
    const int c = p - r * CCOL;

    // tap bounds so that r+3i-12 in [0,54] and c+3j-12 in [0,54]
    const int ilo = (r >= 12) ? 0 : (14 - r) / 3;
    const int ihi = ((66 - r) / 3) < 8 ? ((66 - r) / 3) : 8;
    const int jlo = (c >= 12) ? 0 : (14 - c) / 3;
    const int jhi = ((66 - c) / 3) < 8 ? ((66 - c) / 3) : 8;

    float4 acc = make_float4(0.f, 0.f, 0.f, 0.f);
    for (int i = ilo; i <= ihi; ++i) {
      const int rr2 = r + 3 * i - 12;
      // 8-ch groups: float4 stride of 2 per pixel / per tap
      const float4* __restrict__ xr =
          (const float4*)&sm.x[rr2 * (CCOL * CG) + ch4];
      const float4* __restrict__ kr =
          (const float4*)&sm.k[i * (NT * CG) + ch4];
      for (int j = jlo; j <= jhi; ++j) {
        const int cc2 = c + 3 * j - 12;
        const float4 xv = xr[cc2 * 2];   // ds_load_b128
        const float4 kv = kr[j * 2];     // ds_load_b128
        acc.x = fmaf(xv.x, kv.x, acc.x);
        acc.y = fmaf(xv.y, kv.y, acc.y);
        acc.z = fmaf(xv.z, kv.z, acc.z);
        acc.w = fmaf(xv.w, kv.w, acc.w);
      }
    }

    const float4 xc = *(const float4*)&sm.x[p * CG + ch4];
    float4 o;  // out = lat*x + bias + x
    o.x = fmaf(acc.x, xc.x, xc.x + bv.x);
    o.y = fmaf(acc.y, xc.y, xc.y + bv.y);
    o.z = fmaf(acc.z, xc.z, xc.z + bv.z);
    o.w = fmaf(acc.w, xc.w, xc.w + bv.w);
    *(float4*)&out[obase + (size_t)p * CHN] = o;  // global_store_b128
  }
}

extern "C" void kernel_launch(void* const* d_in, const int* in_sizes, int n_in,
                              void* d_out, int out_size, void* d_ws, size_t ws_size,
                              hipStream_t stream) {
  (void)in_sizes; (void)n_in; (void)out_size; (void)d_ws; (void)ws_size;
  const float* x    = (const float*)d_in[0];  // [64,55,55,96] f32
  const float* rk   = (const float*)d_in[1];  // [96,25,25]    f32
  const float* bias = (const float*)d_in[2];  // [96]          f32
  float*       out  = (float*)d_out;          // [64,55,55,96] f32

  contour_lateral_kernel<<<dim3(NBLK), dim3(256), 0, stream>>>(x, rk, bias, out);
}